// LowRankAttention_31774168056237
// MI455X (gfx1250) — compile-verified
//
#include <hip/hip_runtime.h>

// CDNA5 / gfx1250 fused low-rank-attention mirror-descent kernel.
// One workgroup per (b,h); 10 optimization steps fused in-kernel.
// All S-dimension contractions (RR, KR, LL, LQ, RV) run on
// V_WMMA_F32_16X16X4_F32 per-wave, cross-wave reduced with ds_add_f32.
// WMMA phase is split into two passes (right-side then left-side) to keep
// live v8f accumulators at 40 VGPRs and avoid scratch spills.

typedef __attribute__((ext_vector_type(2))) float v2f;
typedef __attribute__((ext_vector_type(8))) float v8f;

namespace {
constexpr int   Sdim   = 8192;
constexpr int   RANKD  = 16;
constexpr int   Dd     = 64;
constexpr int   NSTEPS = 10;
constexpr float STEPSZ = 1000.0f;
constexpr int   NT     = 512;                 // 16 waves (wave32)
constexpr int   NWAVE  = NT / 32;
constexpr int   SLICE  = Sdim / NWAVE;        // 512 S-elements per wave
constexpr float INV_S2 = 1.0f / ((float)Sdim * (float)Sdim);
}

__device__ __forceinline__ v8f wmma4(v2f a, v2f b, v8f c) {
  // D = A(16x4,f32) * B(4x16,f32) + C(16x16,f32)
  return __builtin_amdgcn_wmma_f32_16x16x4_f32(
      /*neg_a=*/false, a, /*neg_b=*/false, b,
      /*c_mod=*/(short)0, c, /*reuse_a=*/false, /*reuse_b=*/false);
}

__global__ __launch_bounds__(NT) void
lowrank_attn_kernel(const float* __restrict__ Qg, const float* __restrict__ Kg,
                    const float* __restrict__ Vg, float* __restrict__ LPg,
                    float* __restrict__ RPg, float* __restrict__ OUTg)
{
  __shared__ float sLpInv2[Sdim];          // 1/||lp_j||^2, 32KB
  __shared__ float sRR[RANKD * RANKD];     // R R^T
  __shared__ float sLL[RANKD * RANKD];     // L^T L
  __shared__ float sKR[RANKD * Dd];        // R K   (reused as RV at the end)
  __shared__ float sLQ[RANKD * Dd];        // L^T Q
  __shared__ float sRpN2[RANKD];           // ||rp_k||^2
  __shared__ float sC[RANKD];              // tangent correction for rp

  const int bh = blockIdx.x;
  const float* q  = Qg + (size_t)bh * Sdim * Dd;
  const float* kk = Kg + (size_t)bh * Sdim * Dd;
  const float* vv = Vg + (size_t)bh * Sdim * Dd;
  float* lp   = LPg + (size_t)bh * Sdim * RANKD;   // [S, r] row-major
  float* rp   = RPg + (size_t)bh * RANKD * Sdim;   // [r, S] row-major
  float* outp = OUTg + (size_t)bh * Sdim * Dd;

  const int tid  = threadIdx.x;
  const int lane = tid & 31;
  const int wave = tid >> 5;
  const int hl   = lane >> 4;      // lane half: selects K = 2*hl + {0,1}
  const int ln   = lane & 15;      // M (A) / N (B,D) index
  const v8f vz8  = {0.f, 0.f, 0.f, 0.f, 0.f, 0.f, 0.f, 0.f};
  const int base = wave * SLICE;

  for (int step = 0; step < NSTEPS; ++step) {
    // ---- zero LDS accumulators ----
    for (int i = tid; i < RANKD * RANKD; i += NT) { sRR[i] = 0.f; sLL[i] = 0.f; }
    for (int i = tid; i < RANKD * Dd;  i += NT)   { sKR[i] = 0.f; sLQ[i] = 0.f; }
    if (tid < RANKD) { sRpN2[tid] = 0.f; sC[tid] = 0.f; }
    __syncthreads();

    // ---- norms ----
    for (int j = tid; j < Sdim; j += NT) {         // 1/||lp_j||^2 per row
      const float4* r4 = (const float4*)(lp + (size_t)j * RANKD);
      float n = 0.f;
      #pragma unroll
      for (int x = 0; x < RANKD / 4; ++x) {
        float4 t = r4[x];
        n += t.x * t.x + t.y * t.y + t.z * t.z + t.w * t.w;
      }
      sLpInv2[j] = 1.0f / n;
    }
    {                                              // ||rp_k||^2 (reduce over S)
      float a[RANKD];
      #pragma unroll
      for (int x = 0; x < RANKD; ++x) a[x] = 0.f;
      for (int i = tid; i < Sdim; i += NT) {
        #pragma unroll
        for (int x = 0; x < RANKD; ++x) { float t = rp[x * Sdim + i]; a[x] += t * t; }
      }
      #pragma unroll
      for (int x = 0; x < RANKD; ++x) atomicAdd(&sRpN2[x], a[x]);
    }
    __syncthreads();

    // ---- WMMA pass A: RR (Gram, A==B) and KR over this wave's S-slice ----
    {
      const float rsi = 1.0f / sRpN2[ln];          // 1/||rp_m||^2, m = ln
      v8f accRR = vz8;
      v8f accKR[4];
      #pragma unroll
      for (int t = 0; t < 4; ++t) accKR[t] = vz8;
      for (int c0 = 0; c0 < SLICE; c0 += 4) {
        const int i0 = base + c0 + 2 * hl;         // this lane's K-pair {i0, i0+1}
        // A==B chunk of R: R[m, i] = rp[m,i]^2 / ||rp_m||^2
        float r0 = rp[ln * Sdim + i0], r1 = rp[ln * Sdim + i0 + 1];
        v2f aR = { r0 * r0 * rsi, r1 * r1 * rsi };
        accRR = wmma4(aR, aR, accRR);
        #pragma unroll
        for (int t = 0; t < 4; ++t) {
          v2f bK = { kk[(size_t)i0 * Dd + t * 16 + ln],
                     kk[(size_t)(i0 + 1) * Dd + t * 16 + ln] };
          accKR[t] = wmma4(aR, bK, accKR[t]);      // KR[l,a] = sum_i R[l,i] K[i,a]
        }
      }
      // D layout: VGPR v, lanes 0-15 -> M=v, lanes 16-31 -> M=v+8, N=lane&15
      #pragma unroll
      for (int r8 = 0; r8 < 8; ++r8) {
        const int row = r8 + 8 * hl;
        atomicAdd(&sRR[row * RANKD + ln], accRR[r8]);
        #pragma unroll
        for (int t = 0; t < 4; ++t)
          atomicAdd(&sKR[row * Dd + t * 16 + ln], accKR[t][r8]);
      }
    }
    // ---- WMMA pass B: LL (Gram, A==B) and LQ over this wave's S-slice ----
    {
      v8f accLL = vz8;
      v8f accLQ[4];
      #pragma unroll
      for (int t = 0; t < 4; ++t) accLQ[t] = vz8;
      for (int c0 = 0; c0 < SLICE; c0 += 4) {
        const int i0 = base + c0 + 2 * hl;
        // A==B chunk of L^T: L[j, m] = lp[j,m]^2 / ||lp_j||^2
        float l0 = lp[(size_t)i0 * RANKD + ln];
        float l1 = lp[(size_t)(i0 + 1) * RANKD + ln];
        v2f aL = { l0 * l0 * sLpInv2[i0], l1 * l1 * sLpInv2[i0 + 1] };
        accLL = wmma4(aL, aL, accLL);
        #pragma unroll
        for (int t = 0; t < 4; ++t) {
          v2f bQ = { q[(size_t)i0 * Dd + t * 16 + ln],
                     q[(size_t)(i0 + 1) * Dd + t * 16 + ln] };
          accLQ[t] = wmma4(aL, bQ, accLQ[t]);      // LQ[k,a] = sum_j L[j,k] Q[j,a]
        }
      }
      #pragma unroll
      for (int r8 = 0; r8 < 8; ++r8) {
        const int row = r8 + 8 * hl;
        atomicAdd(&sLL[row * RANKD + ln], accLL[r8]);
        #pragma unroll
        for (int t = 0; t < 4; ++t)
          atomicAdd(&sLQ[row * Dd + t * 16 + ln], accLQ[t][r8]);
      }
    }
    __syncthreads();

    // ---- left update (row-parallel; each thread owns disjoint rows) ----
    for (int j = tid; j < Sdim; j += NT) {
      float row[RANKD], ls[RANKD], Lr[RANKD], g[RANKD];
      float n2 = 0.f;
      #pragma unroll
      for (int x = 0; x < RANKD; ++x) { row[x] = lp[(size_t)j * RANKD + x]; n2 += row[x] * row[x]; }
      const float inv = rsqrtf(n2);
      #pragma unroll
      for (int x = 0; x < RANKD; ++x) { ls[x] = row[x] * inv; Lr[x] = ls[x] * ls[x]; }
      #pragma unroll
      for (int l = 0; l < RANKD; ++l) {
        float acc = 0.f;
        #pragma unroll
        for (int x = 0; x < RANKD; ++x) acc += Lr[x] * sRR[x * RANKD + l];
        g[l] = acc;
      }
      const float4* q4 = (const float4*)(q + (size_t)j * Dd);
      for (int a4 = 0; a4 < Dd / 4; ++a4) {
        float4 qa = q4[a4];
        #pragma unroll
        for (int l = 0; l < RANKD; ++l) {
          const float* kr = &sKR[l * Dd + a4 * 4];
          g[l] -= qa.x * kr[0] + qa.y * kr[1] + qa.z * kr[2] + qa.w * kr[3];
        }
      }
      #pragma unroll
      for (int l = 0; l < RANKD; ++l) g[l] *= INV_S2 * 2.0f * ls[l];
      float dot = 0.f;
      #pragma unroll
      for (int x = 0; x < RANKD; ++x) dot += ls[x] * g[x];
      #pragma unroll
      for (int x = 0; x < RANKD; ++x) {
        float d = (g[x] - ls[x] * dot) * inv;      // tangent proj, /||lp||
        lp[(size_t)j * RANKD + x] = row[x] - STEPSZ * d;
      }
    }

    // ---- right update pass 1: accumulate c_k = sum_i rs*t (column-parallel) ----
    {
      float rin[RANKD], cacc[RANKD];
      #pragma unroll
      for (int x = 0; x < RANKD; ++x) { rin[x] = rsqrtf(sRpN2[x]); cacc[x] = 0.f; }
      for (int i = tid; i < Sdim; i += NT) {
        float rs[RANKD], Rc[RANKD], tarr[RANKD];
        #pragma unroll
        for (int l = 0; l < RANKD; ++l) { float t = rp[l * Sdim + i]; rs[l] = t * rin[l]; Rc[l] = rs[l] * rs[l]; }
        #pragma unroll
        for (int k2 = 0; k2 < RANKD; ++k2) {
          float acc = 0.f;
          #pragma unroll
          for (int l = 0; l < RANKD; ++l) acc += sLL[k2 * RANKD + l] * Rc[l];
          tarr[k2] = acc;
        }
        const float4* k4 = (const float4*)(kk + (size_t)i * Dd);
        for (int a4 = 0; a4 < Dd / 4; ++a4) {
          float4 ka = k4[a4];
          #pragma unroll
          for (int k2 = 0; k2 < RANKD; ++k2) {
            const float* lq = &sLQ[k2 * Dd + a4 * 4];
            tarr[k2] -= ka.x * lq[0] + ka.y * lq[1] + ka.z * lq[2] + ka.w * lq[3];
          }
        }
        #pragma unroll
        for (int k2 = 0; k2 < RANKD; ++k2) {
          float t = tarr[k2] * INV_S2 * 2.0f * rs[k2];
          cacc[k2] += rs[k2] * t;
        }
      }
      #pragma unroll
      for (int x = 0; x < RANKD; ++x) atomicAdd(&sC[x], cacc[x]);
    }
    __syncthreads();

    // ---- right update pass 2: apply (recompute t, columns are thread-private) ----
    {
      float rin[RANKD], cc[RANKD];
      #pragma unroll
      for (int x = 0; x < RANKD; ++x) { rin[x] = rsqrtf(sRpN2[x]); cc[x] = sC[x]; }
      for (int i = tid; i < Sdim; i += NT) {
        float rpv[RANKD], rs[RANKD], Rc[RANKD], tarr[RANKD];
        #pragma unroll
        for (int l = 0; l < RANKD; ++l) { rpv[l] = rp[l * Sdim + i]; rs[l] = rpv[l] * rin[l]; Rc[l] = rs[l] * rs[l]; }
        #pragma unroll
        for (int k2 = 0; k2 < RANKD; ++k2) {
          float acc = 0.f;
          #pragma unroll
          for (int l = 0; l < RANKD; ++l) acc += sLL[k2 * RANKD + l] * Rc[l];
          tarr[k2] = acc;
        }
        const float4* k4 = (const float4*)(kk + (size_t)i * Dd);
        for (int a4 = 0; a4 < Dd / 4; ++a4) {
          float4 ka = k4[a4];
          #pragma unroll
          for (int k2 = 0; k2 < RANKD; ++k2) {
            const float* lq = &sLQ[k2 * Dd + a4 * 4];
            tarr[k2] -= ka.x * lq[0] + ka.y * lq[1] + ka.z * lq[2] + ka.w * lq[3];
          }
        }
        #pragma unroll
        for (int k2 = 0; k2 < RANKD; ++k2) {
          float t = tarr[k2] * INV_S2 * 2.0f * rs[k2];
          float d = (t - rs[k2] * cc[k2]) * rin[k2];
          rp[k2 * Sdim + i] = rpv[k2] - STEPSZ * d;
        }
      }
    }
    __syncthreads();
  }

  // ==== final: out = left @ (right @ V) ====
  if (tid < RANKD) sRpN2[tid] = 0.f;
  for (int i = tid; i < RANKD * Dd; i += NT) sKR[i] = 0.f;   // reuse sKR as RV
  __syncthreads();
  {
    float a[RANKD];
    #pragma unroll
    for (int x = 0; x < RANKD; ++x) a[x] = 0.f;
    for (int i = tid; i < Sdim; i += NT) {
      #pragma unroll
      for (int x = 0; x < RANKD; ++x) { float t = rp[x * Sdim + i]; a[x] += t * t; }
    }
    #pragma unroll
    for (int x = 0; x < RANKD; ++x) atomicAdd(&sRpN2[x], a[x]);
  }
  __syncthreads();
  {
    const float rsi = 1.0f / sRpN2[ln];
    v8f accRV[4];
    #pragma unroll
    for (int t = 0; t < 4; ++t) accRV[t] = vz8;
    for (int c0 = 0; c0 < SLICE; c0 += 4) {
      const int i0 = base + c0 + 2 * hl;
      float r0 = rp[ln * Sdim + i0], r1 = rp[ln * Sdim + i0 + 1];
      v2f aR = { r0 * r0 * rsi, r1 * r1 * rsi };
      #pragma unroll
      for (int t = 0; t < 4; ++t) {
        v2f bV = { vv[(size_t)i0 * Dd + t * 16 + ln],
                   vv[(size_t)(i0 + 1) * Dd + t * 16 + ln] };
        accRV[t] = wmma4(aR, bV, accRV[t]);        // RV[k,d] = sum_i right[k,i] V[i,d]
      }
    }
    #pragma unroll
    for (int r8 = 0; r8 < 8; ++r8) {
      const int row = r8 + 8 * hl;
      #pragma unroll
      for (int t = 0; t < 4; ++t)
        atomicAdd(&sKR[row * Dd + t * 16 + ln], accRV[t][r8]);
    }
  }
  __syncthreads();
  for (int j = tid; j < Sdim; j += NT) {
    float row[RANKD];
    float n2 = 0.f;
    #pragma unroll
    for (int x = 0; x < RANKD; ++x) { row[x] = lp[(size_t)j * RANKD + x]; n2 += row[x] * row[x]; }
    const float in2 = 1.0f / n2;
    #pragma unroll
    for (int t4 = 0; t4 < 4; ++t4) {               // 16-wide output tiles, low reg use
      float acc[16];
      #pragma unroll
      for (int a = 0; a < 16; ++a) acc[a] = 0.f;
      #pragma unroll
      for (int x = 0; x < RANKD; ++x) {
        float Lx = row[x] * row[x] * in2;
        #pragma unroll
        for (int a = 0; a < 16; ++a) acc[a] += Lx * sKR[x * Dd + t4 * 16 + a];
      }
      float4* o4 = (float4*)(outp + (size_t)j * Dd + t4 * 16);
      #pragma unroll
      for (int a4 = 0; a4 < 4; ++a4) {
        float4 ov = { acc[a4 * 4 + 0], acc[a4 * 4 + 1], acc[a4 * 4 + 2], acc[a4 * 4 + 3] };
        o4[a4] = ov;
      }
    }
  }
}

extern "C" void kernel_launch(void* const* d_in, const int* in_sizes, int n_in,
                              void* d_out, int out_size, void* d_ws, size_t ws_size,
                              hipStream_t stream) {
  const float* Qg  = (const float*)d_in[0];
  const float* Kg  = (const float*)d_in[1];
  const float* Vg  = (const float*)d_in[2];
  const float* LP0 = (const float*)d_in[3];
  const float* RP0 = (const float*)d_in[4];

  const int nbh = in_sizes[3] / (Sdim * RANKD);    // B*H = 24
  const size_t pelems = (size_t)nbh * Sdim * RANKD;

  // Working copies of the optimized parameters (inputs must not be mutated).
  float* wlp = (float*)d_ws;
  float* wrp = wlp + pelems;
  hipMemcpyAsync(wlp, LP0, pelems * sizeof(float), hipMemcpyDeviceToDevice, stream);
  hipMemcpyAsync(wrp, RP0, pelems * sizeof(float), hipMemcpyDeviceToDevice, stream);

  lowrank_attn_kernel<<<nbh, NT, 0, stream>>>(Qg, Kg, Vg, wlp, wrp, (float*)d_out);
}